// HAN_LP_27917287424411
// MI455X (gfx1250) — compile-verified
//
#include <hip/hip_runtime.h>

typedef __attribute__((ext_vector_type(2))) float v2f;
typedef __attribute__((ext_vector_type(8))) float v8f;

// ---------- helpers: order-preserving float<->uint mapping for atomicMax ----------
__device__ __forceinline__ unsigned f2ord(float f) {
  unsigned u = __float_as_uint(f);
  return (u & 0x80000000u) ? ~u : (u | 0x80000000u);
}
__device__ __forceinline__ float ord2f(unsigned u) {
  return __uint_as_float((u & 0x80000000u) ? (u & 0x7FFFFFFFu) : ~u);
}

// ---------- h[N,Fout] = x[N,128] @ W[128,Fout] + b  via V_WMMA_F32_16X16X4_F32 ----
// All GEMMs in this net have K = 128. 256-thread block = 8 waves.
//   Fout=128: 1 row tile (16 rows) x 8 col tiles per block
//   Fout= 64: 2 row tiles          x 4 col tiles per block
// The x row tile(s) are staged once into LDS (coalesced float4, padded stride to
// kill bank conflicts), then each wave runs the K-loop from LDS (ds_load_b64).
#define GEMM_K 128
#define LDS_STRIDE (GEMM_K + 4)  // 132 floats; 132 mod 64 = 4 -> conflict-free column reads

__global__ __launch_bounds__(256) void gemm_bias_wmma(
    const float* __restrict__ x, const float* __restrict__ W,
    const float* __restrict__ b, float* __restrict__ h, int N, int Fout) {
  __shared__ float sx[32 * LDS_STRIDE];  // up to 2 row tiles of 16 rows x 128 cols

  const int wave = threadIdx.x >> 5;
  const int lane = threadIdx.x & 31;
  const int half = lane >> 4;  // 0: lanes 0-15, 1: lanes 16-31
  const int r    = lane & 15;
  const int tilesPerRow    = Fout >> 4;        // 8 or 4
  const int rowTilesPerBlk = 8 / tilesPerRow;  // 1 or 2
  const int blockRow0 = (int)blockIdx.x * rowTilesPerBlk * 16;
  const int localRowTile = wave / tilesPerRow;
  const int colTile      = wave % tilesPerRow;
  const int m0 = blockRow0 + localRowTile * 16;
  const int n0 = colTile << 4;

  // ---- cooperative stage of rowTilesPerBlk*16 rows of x into LDS ----
  const int totalRows = rowTilesPerBlk * 16;
  for (int i = threadIdx.x; i < totalRows * (GEMM_K / 4); i += 256) {
    const int row = i / (GEMM_K / 4);
    const int c4  = i % (GEMM_K / 4);
    int gr = blockRow0 + row;
    if (gr >= N) gr = N - 1;  // clamp (stores are guarded later)
    *(float4*)&sx[row * LDS_STRIDE + c4 * 4] =
        *(const float4*)&x[(long)gr * GEMM_K + c4 * 4];
  }
  __syncthreads();

  if (m0 >= N) return;  // wave-uniform exit (EXEC stays all-1 for WMMA)

  const float* __restrict__ xrow = &sx[(localRowTile * 16 + r) * LDS_STRIDE];
  v8f c = {};
#pragma unroll 4
  for (int k0 = 0; k0 < GEMM_K; k0 += 4) {
    const int ka = k0 + (half << 1);
    v2f a, bb;
    // A 16x4: lanes 0-15 hold K=k0,k0+1 ; lanes 16-31 hold K=k0+2,k0+3
    a.x = xrow[ka];
    a.y = xrow[ka + 1];
    // B 4x16: N across lanes, K split the same way
    bb.x = W[ka * Fout + n0 + r];
    bb.y = W[(ka + 1) * Fout + n0 + r];
    c = __builtin_amdgcn_wmma_f32_16x16x4_f32(false, a, false, bb, (short)0, c,
                                              false, false);
  }
  const float bias = b[n0 + r];
#pragma unroll
  for (int i = 0; i < 8; ++i) {
    const int m = m0 + i + (half << 3);  // C/D layout: VGPR i -> M = i (+8 for hi half)
    if (m < N) h[(long)m * Fout + n0 + r] = c[i] + bias;
  }
}

// ---------- alpha[n*H+h] = sum_d h[n*F + h*D + d] * p[h*D + d] ----------
__global__ void node_alpha(const float* __restrict__ hfeat, const float* __restrict__ p,
                           float* __restrict__ alpha, long N, int F, int D, int H) {
  long id = (long)blockIdx.x * blockDim.x + threadIdx.x;  // over N*H
  if (id >= N * H) return;
  const int hh = (int)(id % H);
  const long n = id / H;
  const float* __restrict__ row = hfeat + n * F + hh * D;
  const float* __restrict__ pp = p + hh * D;
  float s = 0.f;
  for (int d = 0; d < D; d += 4) {
    s += row[d] * pp[d] + row[d + 1] * pp[d + 1] + row[d + 2] * pp[d + 2] +
         row[d + 3] * pp[d + 3];
  }
  alpha[id] = s;
}

__global__ void fill_u32(unsigned* __restrict__ p, unsigned v, long n) {
  long id = (long)blockIdx.x * blockDim.x + threadIdx.x;
  if (id < n) p[id] = v;
}

// ---------- pass 1: per-(dst,h) segment max of leaky_relu(alpha) ----------
__global__ void edge_max(const int* __restrict__ src, const int* __restrict__ dst,
                         const float* __restrict__ as, const float* __restrict__ ad,
                         unsigned* __restrict__ amax, long E, int H) {
  long id = (long)blockIdx.x * blockDim.x + threadIdx.x;  // over E*H
  if (id >= E * H) return;
  const long e = id / H;
  const int hh = (int)(id % H);
  float a = as[(long)src[e] * H + hh] + ad[(long)dst[e] * H + hh];
  a = (a >= 0.f) ? a : 0.2f * a;  // leaky relu
  atomicMax(&amax[(long)dst[e] * H + hh], f2ord(a));
}

// ---------- pass 2: ex = exp(alpha - amax[dst]); denom[dst] += ex ----------
__global__ void edge_softmax(const int* __restrict__ src, const int* __restrict__ dst,
                             const float* __restrict__ as, const float* __restrict__ ad,
                             const unsigned* __restrict__ amax, float* __restrict__ ex,
                             float* __restrict__ denom, long E, int H) {
  long id = (long)blockIdx.x * blockDim.x + threadIdx.x;  // over E*H
  if (id >= E * H) return;
  const long e = id / H;
  const int hh = (int)(id % H);
  const long dj = (long)dst[e];
  float a = as[(long)src[e] * H + hh] + ad[dj * H + hh];
  a = (a >= 0.f) ? a : 0.2f * a;
  const float m = ord2f(amax[dj * H + hh]);  // finite for any dst that has edges
  const float v = expf(a - m);
  ex[id] = v;
  atomicAdd(&denom[dj * H + hh], v);
}

// ---------- pass 3: out[dst] += h_src[src] * ex/(denom+eps), float4 chunks ----------
__global__ void edge_aggr(const int* __restrict__ src, const int* __restrict__ dst,
                          const float* __restrict__ hsrc, const float* __restrict__ ex,
                          const float* __restrict__ denom, float* __restrict__ out,
                          long E, int F, int D, int H) {
  const int chunks = F >> 2;
  long id = (long)blockIdx.x * blockDim.x + threadIdx.x;  // over E*chunks
  if (id >= E * chunks) return;
  const long e = id / chunks;
  const int c = (int)(id % chunks);
  const int f0 = c << 2;
  const int hh = f0 / D;  // D is a multiple of 4 -> chunk stays inside one head
  const long sj = (long)src[e];
  const long dj = (long)dst[e];
  const float w = ex[e * H + hh] / (denom[dj * H + hh] + 1e-16f);
  const float4 v = *(const float4*)(hsrc + sj * F + f0);
  float* o = out + dj * F + f0;
  atomicAdd(o + 0, v.x * w);
  atomicAdd(o + 1, v.y * w);
  atomicAdd(o + 2, v.z * w);
  atomicAdd(o + 3, v.w * w);
}

__global__ void relu_inplace(float* __restrict__ p, long n) {
  long id = (long)blockIdx.x * blockDim.x + threadIdx.x;
  if (id < n) p[id] = fmaxf(p[id], 0.f);
}

// ---------- decoder: out[l] = dot(z_p[src_l], z_a[dst_l]) ----------
__global__ void predict(const float* __restrict__ zp, const float* __restrict__ za,
                        const int* __restrict__ esrc, const int* __restrict__ edst,
                        float* __restrict__ out, long L, int F) {
  long l = (long)blockIdx.x * blockDim.x + threadIdx.x;
  if (l >= L) return;
  const float* a = zp + (long)esrc[l] * F;
  const float* b = za + (long)edst[l] * F;
  float s = 0.f;
  for (int f = 0; f < F; f += 4) {
    const float4 va = *(const float4*)(a + f);
    const float4 vb = *(const float4*)(b + f);
    s += va.x * vb.x + va.y * vb.y + va.z * vb.z + va.w * vb.w;
  }
  out[l] = s;
}

// =============================== host launcher ===============================
extern "C" void kernel_launch(void* const* d_in, const int* in_sizes, int n_in,
                              void* d_out, int out_size, void* d_ws, size_t ws_size,
                              hipStream_t stream) {
  const float* x_p  = (const float*)d_in[0];
  const float* x_a  = (const float*)d_in[1];
  const int*   ei_pa = (const int*)d_in[2];
  const int*   ei_ap = (const int*)d_in[3];
  const int*   eli   = (const int*)d_in[4];
  const float* p1W = (const float*)d_in[5];
  const float* p1b = (const float*)d_in[6];
  const float* a1W = (const float*)d_in[7];
  const float* a1b = (const float*)d_in[8];
  const float* s1pa = (const float*)d_in[9];
  const float* d1pa = (const float*)d_in[10];
  const float* s1ap = (const float*)d_in[11];
  const float* d1ap = (const float*)d_in[12];
  // d_in[13..15] = k1W,k1b,q1 : dead (softmax over a single metapath == identity)
  const float* p2W = (const float*)d_in[16];
  const float* p2b = (const float*)d_in[17];
  const float* a2W = (const float*)d_in[18];
  const float* a2b = (const float*)d_in[19];
  const float* s2pa = (const float*)d_in[20];
  const float* d2pa = (const float*)d_in[21];
  const float* s2ap = (const float*)d_in[22];
  const float* d2ap = (const float*)d_in[23];
  // d_in[24..26] = k2W,k2b,q2 : dead

  const int F_IN = 128, F1 = 128, F2 = 64, H = 4, D1 = F1 / H, D2 = F2 / H;
  (void)F_IN;
  const long NP = in_sizes[0] / 128;
  const long NA = in_sizes[1] / 128;
  const long E  = in_sizes[2] / 2;
  const long L  = in_sizes[4] / 2;
  const long NMAX = (NP > NA) ? NP : NA;

  // ---- workspace layout (floats) ----
  float* w = (float*)d_ws;
  auto take = [&](long n) { float* p = w; w += n; return p; };
  float* h_p1 = take(NP * F1);
  float* h_a1 = take(NA * F1);
  float* z_p1 = take(NP * F1);
  float* z_a1 = take(NA * F1);
  float* h_p2 = take(NP * F2);
  float* h_a2 = take(NA * F2);
  float* z_p2 = take(NP * F2);
  float* z_a2 = take(NA * F2);
  float* as_pa = take(NP * H);
  float* ad_pa = take(NA * H);
  float* as_ap = take(NA * H);
  float* ad_ap = take(NP * H);
  unsigned* amax = (unsigned*)take(NMAX * H);
  float* denom = take(NMAX * H);
  float* ex = take(E * H);
  (void)ws_size; (void)n_in; (void)out_size;

  const dim3 blk(256);
  auto cdiv = [](long a, long b) { return (unsigned)((a + b - 1) / b); };

  auto gemm = [&](const float* x, const float* W_, const float* b_, float* hout,
                  long N, int Fout) {
    const int rowTilesPerBlk = 8 / (Fout >> 4);  // 1 (Fout=128) or 2 (Fout=64)
    const long rowTiles = (N + 15) / 16;
    gemm_bias_wmma<<<cdiv(rowTiles, rowTilesPerBlk), blk, 0, stream>>>(
        x, W_, b_, hout, (int)N, Fout);
  };

  auto metapath = [&](const int* e2, const float* hsrc, const float* as,
                      const float* ad, float* out, long Ndst, int F, int D) {
    const int* src = e2;       // row 0 of [2,E]
    const int* dst = e2 + E;   // row 1 of [2,E]
    const long nh = Ndst * H;
    fill_u32<<<cdiv(nh, 256), blk, 0, stream>>>(amax, 0x007FFFFFu, nh);  // f2ord(-inf)
    fill_u32<<<cdiv(nh, 256), blk, 0, stream>>>((unsigned*)denom, 0u, nh);
    fill_u32<<<cdiv(Ndst * (long)F, 256), blk, 0, stream>>>((unsigned*)out, 0u,
                                                            Ndst * (long)F);
    edge_max<<<cdiv(E * H, 256), blk, 0, stream>>>(src, dst, as, ad, amax, E, H);
    edge_softmax<<<cdiv(E * H, 256), blk, 0, stream>>>(src, dst, as, ad, amax, ex,
                                                       denom, E, H);
    edge_aggr<<<cdiv(E * (long)(F >> 2), 256), blk, 0, stream>>>(src, dst, hsrc, ex,
                                                                 denom, out, E, F, D, H);
    relu_inplace<<<cdiv(Ndst * (long)F, 256), blk, 0, stream>>>(out, Ndst * (long)F);
  };

  // ================= layer 1 =================
  gemm(x_p, p1W, p1b, h_p1, NP, F1);
  gemm(x_a, a1W, a1b, h_a1, NA, F1);
  node_alpha<<<cdiv(NP * H, 256), blk, 0, stream>>>(h_p1, s1pa, as_pa, NP, F1, D1, H);
  node_alpha<<<cdiv(NA * H, 256), blk, 0, stream>>>(h_a1, d1pa, ad_pa, NA, F1, D1, H);
  node_alpha<<<cdiv(NA * H, 256), blk, 0, stream>>>(h_a1, s1ap, as_ap, NA, F1, D1, H);
  node_alpha<<<cdiv(NP * H, 256), blk, 0, stream>>>(h_p1, d1ap, ad_ap, NP, F1, D1, H);
  metapath(ei_pa, h_p1, as_pa, ad_pa, z_a1, NA, F1, D1);  // papers -> authors
  metapath(ei_ap, h_a1, as_ap, ad_ap, z_p1, NP, F1, D1);  // authors -> papers

  // ================= layer 2 =================
  gemm(z_p1, p2W, p2b, h_p2, NP, F2);
  gemm(z_a1, a2W, a2b, h_a2, NA, F2);
  node_alpha<<<cdiv(NP * H, 256), blk, 0, stream>>>(h_p2, s2pa, as_pa, NP, F2, D2, H);
  node_alpha<<<cdiv(NA * H, 256), blk, 0, stream>>>(h_a2, d2pa, ad_pa, NA, F2, D2, H);
  node_alpha<<<cdiv(NA * H, 256), blk, 0, stream>>>(h_a2, s2ap, as_ap, NA, F2, D2, H);
  node_alpha<<<cdiv(NP * H, 256), blk, 0, stream>>>(h_p2, d2ap, ad_ap, NP, F2, D2, H);
  metapath(ei_pa, h_p2, as_pa, ad_pa, z_a2, NA, F2, D2);
  metapath(ei_ap, h_a2, as_ap, ad_ap, z_p2, NP, F2, D2);

  // ================= decoder =================
  predict<<<cdiv(L, 256), blk, 0, stream>>>(z_p2, z_a2, eli, eli + L, (float*)d_out,
                                            L, F2);
}